// TripletLoss_79989470921085
// MI455X (gfx1250) — compile-verified
//
#include <hip/hip_runtime.h>
#include <math.h>

#define NROWS 4096
#define DIMS  512
#define NBLK32 128            // NROWS / 32
#define MARGIN_F 2.0f
#define EPS_F 1e-11f
#define TINY_F 1.17549435e-38f

typedef __attribute__((ext_vector_type(2))) float v2f;
typedef __attribute__((ext_vector_type(8))) float v8f;

// ---------------------------------------------------------------------------
// Kernel 1: per-row squared norms. One wave32 per row.
// ---------------------------------------------------------------------------
__global__ void rowsq_kernel(const float* __restrict__ F, float* __restrict__ sq) {
    int row  = (blockIdx.x * blockDim.x + threadIdx.x) >> 5;
    int lane = threadIdx.x & 31;
    if (row >= NROWS) return;
    const float* r = F + (size_t)row * DIMS;
    float s = 0.f;
    #pragma unroll 4
    for (int c = lane; c < DIMS; c += 32) { float v = r[c]; s += v * v; }
    for (int off = 16; off > 0; off >>= 1) s += __shfl_xor(s, off, 32);
    if (lane == 0) sq[row] = s;
}

// ---------------------------------------------------------------------------
// JAX threefry2x32, key = PRNGKey(1) = (k0=0, k1=1).
// ---------------------------------------------------------------------------
__device__ __forceinline__ void threefry2x32(unsigned k0, unsigned k1,
                                             unsigned c0, unsigned c1,
                                             unsigned& o0, unsigned& o1) {
    unsigned ks2 = k0 ^ k1 ^ 0x1BD11BDAu;
    unsigned x0 = c0 + k0, x1 = c1 + k1;
#define TF_RND(r) { x0 += x1; x1 = (x1 << (r)) | (x1 >> (32 - (r))); x1 ^= x0; }
    TF_RND(13) TF_RND(15) TF_RND(26) TF_RND(6)   x0 += k1;  x1 += ks2 + 1u;
    TF_RND(17) TF_RND(29) TF_RND(16) TF_RND(24)  x0 += ks2; x1 += k0  + 2u;
    TF_RND(13) TF_RND(15) TF_RND(26) TF_RND(6)   x0 += k0;  x1 += k1  + 3u;
    TF_RND(17) TF_RND(29) TF_RND(16) TF_RND(24)  x0 += k1;  x1 += ks2 + 4u;
    TF_RND(13) TF_RND(15) TF_RND(26) TF_RND(6)   x0 += ks2; x1 += k0  + 5u;
#undef TF_RND
    o0 = x0; o1 = x1;
}

// Gumbel noise at flat index m of a (NROWS*NROWS) draw, matching
// jax.random.gumbel(key(1), (N,N)): counters split [0..H) x0, [H..2H) x1.
__device__ __forceinline__ float gumbel_at(unsigned m) {
    const unsigned HALF = (unsigned)NROWS * (unsigned)NROWS / 2u;  // 8388608
    unsigned o0, o1, bits;
    if (m < HALF) { threefry2x32(0u, 1u, m, m + HALF, o0, o1); bits = o0; }
    else          { threefry2x32(0u, 1u, m - HALF, m, o0, o1); bits = o1; }
    unsigned ub = (bits >> 9) | 0x3f800000u;       // [1,2) mantissa fill
    float f = __uint_as_float(ub) - 1.0f;          // uniform [0,1)
    float u = fmaxf(f + TINY_F, TINY_F);           // jax: minval=tiny, maxval=1
    return -logf(-logf(u));
}

// ---------------------------------------------------------------------------
// Kernel 2: per-anchor negative via Gumbel-argmax over neg_mask.
// One 256-thread block per row; first-occurrence argmax semantics.
// ---------------------------------------------------------------------------
__global__ void negsample_kernel(const int* __restrict__ labels,
                                 int* __restrict__ neg_idx,
                                 int* __restrict__ has_neg) {
    int i  = blockIdx.x;
    int li = labels[i];
    int t  = threadIdx.x;
    float best = -INFINITY; int bidx = 0; int any = 0;
    for (int j = t; j < NROWS; j += 256) {
        if (labels[j] != li) {
            any = 1;
            float g = gumbel_at((unsigned)i * (unsigned)NROWS + (unsigned)j);
            if (g > best) { best = g; bidx = j; }   // strict > keeps first max
        }
    }
    __shared__ float sval[256];
    __shared__ int   sidx[256];
    __shared__ int   sany[256];
    sval[t] = best; sidx[t] = bidx; sany[t] = any;
    __syncthreads();
    for (int off = 128; off > 0; off >>= 1) {
        if (t < off) {
            float v = sval[t + off]; int ix = sidx[t + off];
            if (v > sval[t] || (v == sval[t] && ix < sidx[t])) { sval[t] = v; sidx[t] = ix; }
            sany[t] |= sany[t + off];
        }
        __syncthreads();
    }
    if (t == 0) { neg_idx[i] = sidx[0]; has_neg[i] = sany[0]; }
}

// ---------------------------------------------------------------------------
// Kernel 3: neg_dist[i] = sqrt(max(sq_i + sq_n - 2 <f_i, f_n>, eps)).
// One wave32 per row.
// ---------------------------------------------------------------------------
__global__ void negdist_kernel(const float* __restrict__ F, const float* __restrict__ sq,
                               const int* __restrict__ neg_idx,
                               float* __restrict__ neg_dist) {
    int row  = (blockIdx.x * blockDim.x + threadIdx.x) >> 5;
    int lane = threadIdx.x & 31;
    if (row >= NROWS) return;
    int j = neg_idx[row];
    const float* a = F + (size_t)row * DIMS;
    const float* b = F + (size_t)j   * DIMS;
    float s = 0.f;
    #pragma unroll 4
    for (int c = lane; c < DIMS; c += 32) s += a[c] * b[c];
    for (int off = 16; off > 0; off >>= 1) s += __shfl_xor(s, off, 32);
    if (lane == 0) {
        float d2 = sq[row] + sq[j] - 2.0f * s;
        neg_dist[row] = sqrtf(fmaxf(d2, EPS_F));
    }
}

// ---------------------------------------------------------------------------
// Kernel 4: fused fp32 WMMA Gram + margin-loss reduction, 2x2 register
// blocking. One wave32 computes a 32x32 output block = four 16x16
// V_WMMA_F32_16X16X4_F32 accumulators; per k-step it loads 2 A-frags +
// 2 B-frags (4x global_load_b64) and issues 4 WMMAs (load:math = 1:1).
//   A-frag lane layout: lane L holds F[rowBase+(L&15)][k + 2*(L>>4) + {0,1}]
//   B-frag lane layout: lane L holds F[colBase+(L&15)][k + 2*(L>>4) + {0,1}]
//   D layout: VGPR v -> (i = rowBase + v + 8*(L>>4), j = colBase + (L&15))
// Only block-upper-triangle (colBlk >= rowBlk) waves run; the exit is
// wave-uniform so EXEC is all 1s at every WMMA. Exact j>i masking in the
// epilogue keeps diagonal blocks correct.
// ---------------------------------------------------------------------------
__global__ void __launch_bounds__(256)
tile_loss_kernel(const float* __restrict__ F, const float* __restrict__ sq,
                 const int* __restrict__ labels, const float* __restrict__ neg_dist,
                 const int* __restrict__ has_neg,
                 double* __restrict__ total, unsigned long long* __restrict__ counter) {
    int wave = threadIdx.x >> 5;
    int lane = threadIdx.x & 31;
    int rowBlk = blockIdx.y;
    int colBlk = blockIdx.x * 8 + wave;
    if (colBlk < rowBlk) return;                   // block upper triangle (uniform)

    int rowBase = rowBlk * 32;
    int colBase = colBlk * 32;
    int half = lane >> 4;                          // 0: K lo pair, 1: K hi pair
    int mrow = lane & 15;

    const float* a0 = F + (size_t)(rowBase + mrow)      * DIMS + 2 * half;
    const float* a1 = F + (size_t)(rowBase + 16 + mrow) * DIMS + 2 * half;
    const float* b0 = F + (size_t)(colBase + mrow)      * DIMS + 2 * half;
    const float* b1 = F + (size_t)(colBase + 16 + mrow) * DIMS + 2 * half;

    v8f acc00 = {}, acc01 = {}, acc10 = {}, acc11 = {};
    #pragma unroll 4
    for (int k = 0; k < DIMS; k += 4) {
        v2f fa0 = *(const v2f*)(a0 + k);
        v2f fa1 = *(const v2f*)(a1 + k);
        v2f fb0 = *(const v2f*)(b0 + k);
        v2f fb1 = *(const v2f*)(b1 + k);
        acc00 = __builtin_amdgcn_wmma_f32_16x16x4_f32(false, fa0, false, fb0,
                                                      (short)0, acc00, false, false);
        acc01 = __builtin_amdgcn_wmma_f32_16x16x4_f32(false, fa0, false, fb1,
                                                      (short)0, acc01, false, false);
        acc10 = __builtin_amdgcn_wmma_f32_16x16x4_f32(false, fa1, false, fb0,
                                                      (short)0, acc10, false, false);
        acc11 = __builtin_amdgcn_wmma_f32_16x16x4_f32(false, fa1, false, fb1,
                                                      (short)0, acc11, false, false);
    }

    int   j0   = colBase + mrow;
    int   j1   = j0 + 16;
    int   lj0  = labels[j0];
    int   lj1  = labels[j1];
    float sqj0 = sq[j0];
    float sqj1 = sq[j1];

    float lsum = 0.f;
    unsigned lcnt = 0;
#define TL_PAIR(ii, jj, sqi, li, nd, g)                                   \
    if ((jj) > (ii) && (li) == labels[jj]) {                              \
        float d2   = (sqi) + ((jj) == j0 ? sqj0 : sqj1) - 2.0f * (g);     \
        float dist = sqrtf(fmaxf(d2, EPS_F));                             \
        lsum += fmaxf(MARGIN_F + dist - (nd), 0.f);                       \
        lcnt += 1u;                                                       \
    }
    #pragma unroll
    for (int v = 0; v < 8; v++) {
        int   i0  = rowBase + v + 8 * half;
        int   i1  = i0 + 16;
        int   li0 = labels[i0];
        int   li1 = labels[i1];
        float sq0 = sq[i0];
        float sq1 = sq[i1];
        if (has_neg[i0]) {
            float nd0 = neg_dist[i0];
            if (j0 > i0 && li0 == lj0) {
                float dist = sqrtf(fmaxf(sq0 + sqj0 - 2.0f * acc00[v], EPS_F));
                lsum += fmaxf(MARGIN_F + dist - nd0, 0.f); lcnt++;
            }
            if (j1 > i0 && li0 == lj1) {
                float dist = sqrtf(fmaxf(sq0 + sqj1 - 2.0f * acc01[v], EPS_F));
                lsum += fmaxf(MARGIN_F + dist - nd0, 0.f); lcnt++;
            }
        }
        if (has_neg[i1]) {
            float nd1 = neg_dist[i1];
            if (j0 > i1 && li1 == lj0) {
                float dist = sqrtf(fmaxf(sq1 + sqj0 - 2.0f * acc10[v], EPS_F));
                lsum += fmaxf(MARGIN_F + dist - nd1, 0.f); lcnt++;
            }
            if (j1 > i1 && li1 == lj1) {
                float dist = sqrtf(fmaxf(sq1 + sqj1 - 2.0f * acc11[v], EPS_F));
                lsum += fmaxf(MARGIN_F + dist - nd1, 0.f); lcnt++;
            }
        }
    }
#undef TL_PAIR
    for (int off = 16; off > 0; off >>= 1) {
        lsum += __shfl_xor(lsum, off, 32);
        lcnt += (unsigned)__shfl_xor((int)lcnt, off, 32);
    }
    if (lane == 0 && lcnt != 0) {
        atomicAdd(total, (double)lsum);
        atomicAdd(counter, (unsigned long long)lcnt);
    }
}

// ---------------------------------------------------------------------------
// Accumulator init + finalize (graph-safe, no memset/sync APIs).
// ---------------------------------------------------------------------------
__global__ void init_acc_kernel(double* total, unsigned long long* counter) {
    *total = 0.0;
    *counter = 0ull;
}

__global__ void finalize_kernel(const double* total, const unsigned long long* counter,
                                float* out) {
    unsigned long long c = *counter;
    out[0] = (c > 0ull) ? (float)(*total / (double)c) : 0.0f;
}

// ---------------------------------------------------------------------------
// Launcher
// ---------------------------------------------------------------------------
extern "C" void kernel_launch(void* const* d_in, const int* in_sizes, int n_in,
                              void* d_out, int out_size, void* d_ws, size_t ws_size,
                              hipStream_t stream) {
    const float* F      = (const float*)d_in[0];   // 4096 x 512 fp32
    const int*   labels = (const int*)d_in[1];     // 4096 int32
    float*       out    = (float*)d_out;           // scalar fp32

    // Workspace layout (needs 64 KiB + 16 B)
    float* sq        = (float*)d_ws;                                   // 4096 f32
    float* neg_dist  = sq + NROWS;                                     // 4096 f32
    int*   neg_idx   = (int*)(neg_dist + NROWS);                       // 4096 i32
    int*   has_neg   = neg_idx + NROWS;                                // 4096 i32
    double* total    = (double*)(has_neg + NROWS);                     // 8B (8-aligned)
    unsigned long long* counter = (unsigned long long*)(total + 1);    // 8B

    rowsq_kernel<<<NROWS / 8, 256, 0, stream>>>(F, sq);
    negsample_kernel<<<NROWS, 256, 0, stream>>>(labels, neg_idx, has_neg);
    negdist_kernel<<<NROWS / 8, 256, 0, stream>>>(F, sq, neg_idx, neg_dist);
    init_acc_kernel<<<1, 1, 0, stream>>>(total, counter);

    dim3 grid(NBLK32 / 8, NBLK32, 1);              // 8 waves/block = 8 col blocks
    tile_loss_kernel<<<grid, 256, 0, stream>>>(F, sq, labels, neg_dist, has_neg,
                                               total, counter);
    finalize_kernel<<<1, 1, 0, stream>>>(total, counter, out);
}